// RL_GAT_Actor_52956946759735
// MI455X (gfx1250) — compile-verified
//
#include <hip/hip_runtime.h>
#include <hip/hip_bf16.h>

#define NN   10000
#define EE   320000
#define EAT  330000      // EE + NN self loops
#define HIDC 256
#define LNEPS 1e-5f

typedef __attribute__((ext_vector_type(2))) float     v2f;
typedef __attribute__((ext_vector_type(8))) float     v8f;
typedef __attribute__((ext_vector_type(4))) unsigned  u32x4;
typedef __attribute__((ext_vector_type(8))) unsigned  u32x8;

// ---------- helpers ----------
__device__ __forceinline__ unsigned fmap(float f) {
    unsigned u = __float_as_uint(f);
    return (u & 0x80000000u) ? ~u : (u | 0x80000000u);
}
__device__ __forceinline__ float funmap(unsigned u) {
    u = (u & 0x80000000u) ? (u & 0x7FFFFFFFu) : ~u;
    return __uint_as_float(u);
}

// Issue a TDM 2D tile load: global (row-major, f32) -> LDS, via D# descriptor
// groups in SGPRs (CDNA5 ISA). Tracked on TENSORcnt.
__device__ __forceinline__ void tdm_load_2d(const float* gptr, unsigned lds_byte,
                                            unsigned row_len, unsigned num_rows,
                                            unsigned tile_w, unsigned tile_h,
                                            unsigned stride_elems) {
    unsigned long long ga = (unsigned long long)(size_t)gptr;
    u32x4 g0;
    g0[0] = 1u;                                             // count=1, user mode
    g0[1] = lds_byte;                                       // lds_addr
    g0[2] = (unsigned)(ga & 0xFFFFFFFFull);                 // global_addr[95:64]
    g0[3] = (unsigned)((ga >> 32) & 0x01FFFFFFull) | (2u << 30); // addr msbs | type=2
    // group1: data_size=4B (2), tensor_dim0=row_len, tensor_dim1=num_rows,
    // tile_dim0=tile_w, tile_dim1=tile_h, tensor_dim0_stride=stride_elems
    unsigned long long q0 = (2ull << 16) | ((unsigned long long)(row_len & 0xFFFFu) << 48);
    unsigned long long q1 = ((unsigned long long)(row_len >> 16) & 0xFFFFull)
                          | ((unsigned long long)num_rows << 16)
                          | ((unsigned long long)(tile_w & 0xFFFFu) << 48);
    unsigned long long q2 = (unsigned long long)(tile_h & 0xFFFFu)
                          | ((unsigned long long)stride_elems << 32);
    u32x8 g1;
    g1[0] = (unsigned)q0; g1[1] = (unsigned)(q0 >> 32);
    g1[2] = (unsigned)q1; g1[3] = (unsigned)(q1 >> 32);
    g1[4] = (unsigned)q2; g1[5] = (unsigned)(q2 >> 32);
    g1[6] = 0u;           g1[7] = 0u;
    asm volatile("tensor_load_to_lds %0, %1" :: "s"(g0), "s"(g1) : "memory");
}

// ---------- zero fill ----------
__global__ void k_fill0(float* p, int n) {
    int i = blockIdx.x * blockDim.x + threadIdx.x;
    if (i < n) p[i] = 0.0f;
}

// ---------- degree + self-loop attr (fill_value='mean') ----------
__global__ void k_deg_loop(const int* __restrict__ dst, const float* __restrict__ ea,
                           float* __restrict__ deg, float* __restrict__ lsum) {
    int e = blockIdx.x * blockDim.x + threadIdx.x;
    if (e >= EE) return;
    int d = dst[e];
    atomicAdd(&deg[d], 1.0f);
    atomicAdd(&lsum[d], ea[e]);
}
__global__ void k_loop_norm(float* __restrict__ lsum, const float* __restrict__ deg) {
    int i = blockIdx.x * blockDim.x + threadIdx.x;
    if (i < NN) lsum[i] /= fmaxf(deg[i], 1.0f);
}

// ---------- dense GEMM: double-buffered TDM -> LDS -> V_WMMA_F32_16X16X4_F32 ----------
// Block of (COLS/32) waves computes a 16 x COLS strip: out = A[NNxK] @ B[KxCOLS] + bias.
// Ping-pong pipeline: wave0 issues TDM loads for chunk c+1 into the alternate LDS
// buffer, waits TENSORcnt<=2 (in-order per wave => chunk c's A+B pair complete),
// barrier publishes, all waves run the fully-unrolled WMMA chunk while c+1 streams in.
// grid.x = NN/16, blockDim.x = COLS (256 -> 8 waves, 128 -> 4 waves).
template<int K, int COLS>
__global__ void k_gemm_wmma(const float* __restrict__ A, int lda,
                            const float* __restrict__ B,
                            const float* __restrict__ bias,
                            float* __restrict__ C, int relu) {
    constexpr int KC  = (K < 16) ? K : 16;
    constexpr int NCH = K / KC;
    __shared__ float ldsA[2][16 * KC];
    __shared__ float ldsB[2][KC * COLS];
    const int rowBase = blockIdx.x * 16;
    const int wave = threadIdx.x >> 5;
    const int lane = threadIdx.x & 31;
    const int half = lane >> 4;          // second half-wave: K+2 / M+8
    const int l16  = lane & 15;
    const int colBase = wave * 32;
    v8f acc0 = {}, acc1 = {};

    if (wave == 0) {                      // prologue: stage chunk 0 into buffer 0
        tdm_load_2d(A + (size_t)rowBase * lda, (unsigned)(size_t)&ldsA[0][0],
                    (unsigned)lda, NN, KC, 16u, (unsigned)lda);
        tdm_load_2d(B, (unsigned)(size_t)&ldsB[0][0],
                    COLS, K, COLS, KC, COLS);
    }
    for (int c = 0; c < NCH; ++c) {
        const int cur = c & 1;
        if (c + 1 < NCH) {
            __syncthreads();              // alternate buffer free (reads from c-1 done)
            if (wave == 0) {
                const int k0 = (c + 1) * KC;
                tdm_load_2d(A + (size_t)rowBase * lda + k0,
                            (unsigned)(size_t)&ldsA[cur ^ 1][0],
                            (unsigned)lda, NN, KC, 16u, (unsigned)lda);
                tdm_load_2d(B + (size_t)k0 * COLS,
                            (unsigned)(size_t)&ldsB[cur ^ 1][0],
                            COLS, K, COLS, KC, COLS);
            }
        }
        if (wave == 0) {
            if (c + 1 < NCH) __builtin_amdgcn_s_wait_tensorcnt(2); // chunk c landed
            else             __builtin_amdgcn_s_wait_tensorcnt(0); // drain
        }
        __syncthreads();                  // publish buffer[cur]
        #pragma unroll
        for (int kk = 0; kk < KC; kk += 4) {
            const int k = kk + 2 * half;
            v2f a, b0, b1;
            a.x  = ldsA[cur][l16 * KC + k];
            a.y  = ldsA[cur][l16 * KC + k + 1];
            b0.x = ldsB[cur][k * COLS + colBase + l16];
            b0.y = ldsB[cur][(k + 1) * COLS + colBase + l16];
            b1.x = ldsB[cur][k * COLS + colBase + 16 + l16];
            b1.y = ldsB[cur][(k + 1) * COLS + colBase + 16 + l16];
            acc0 = __builtin_amdgcn_wmma_f32_16x16x4_f32(false, a, false, b0,
                                                         (short)0, acc0, false, false);
            acc1 = __builtin_amdgcn_wmma_f32_16x16x4_f32(false, a, false, b1,
                                                         (short)0, acc1, false, false);
        }
    }
    const int col0 = colBase + l16;
    const int col1 = colBase + 16 + l16;
    const float bc0 = bias[col0];
    const float bc1 = bias[col1];
    #pragma unroll
    for (int r = 0; r < 8; ++r) {
        const int row = rowBase + r + 8 * half;
        float v0 = acc0[r] + bc0;
        float v1 = acc1[r] + bc1;
        if (relu) { v0 = fmaxf(v0, 0.0f); v1 = fmaxf(v1, 0.0f); }
        C[(size_t)row * COLS + col0] = v0;
        C[(size_t)row * COLS + col1] = v1;
    }
}

// ---------- GATv2 edge phase ----------
// wave per edge; lane covers 8 channels. H heads of C=HID/H channels; att flat (H,C)=channel idx.
template<int H, int C>
__global__ void k_edge_logits(const float* __restrict__ xl, const float* __restrict__ xr,
                              const float* __restrict__ We, const int* __restrict__ src,
                              const int* __restrict__ dst, const float* __restrict__ ea,
                              const float* __restrict__ loopa, const float* __restrict__ att,
                              float* __restrict__ elog, unsigned* __restrict__ mxu) {
    int e = blockIdx.x * (blockDim.x >> 5) + (threadIdx.x >> 5);
    if (e >= EAT) return;
    int lane = threadIdx.x & 31;
    int s, d; float a;
    if (e < EE) { s = src[e]; d = dst[e]; a = ea[e]; }
    else        { s = d = e - EE; a = loopa[e - EE]; }
    const float* xls = xl + (size_t)s * HIDC;
    const float* xrd = xr + (size_t)d * HIDC;
    __builtin_prefetch(xls, 0, 3);       // global_prefetch_b8: warm gather rows
    __builtin_prefetch(xrd, 0, 3);
    int cbase = lane * 8;
    int h = cbase / C;
    float logit = 0.0f;
    #pragma unroll
    for (int j = 0; j < 8; ++j) {
        int c = cbase + j;
        float m = xls[c] + xrd[c] + a * We[c];
        m = (m > 0.0f) ? m : 0.2f * m;          // leaky_relu(0.2)
        logit += m * att[c];
    }
    const int G = 32 / H;                        // lanes per head
    #pragma unroll
    for (int off = G >> 1; off >= 1; off >>= 1)
        logit += __shfl_xor(logit, off, 32);
    if ((lane & (G - 1)) == 0) {
        elog[(size_t)e * H + h] = logit;
        atomicMax(&mxu[(size_t)d * H + h], fmap(logit));
    }
}

template<int H>
__global__ void k_edge_exp(const int* __restrict__ dst, float* __restrict__ elog,
                           const unsigned* __restrict__ mxu, float* __restrict__ den) {
    int idx = blockIdx.x * blockDim.x + threadIdx.x;
    if (idx >= EAT * H) return;
    int e = idx / H, h = idx - e * H;
    int d = (e < EE) ? dst[e] : (e - EE);
    float ex = __expf(elog[idx] - funmap(mxu[(size_t)d * H + h]));
    elog[idx] = ex;
    atomicAdd(&den[(size_t)d * H + h], ex);
}

template<int H, int C>
__global__ void k_edge_scatter(const float* __restrict__ xl, const int* __restrict__ src,
                               const int* __restrict__ dst, const float* __restrict__ elog,
                               const float* __restrict__ den, float* __restrict__ acc) {
    int e = blockIdx.x * (blockDim.x >> 5) + (threadIdx.x >> 5);
    if (e >= EAT) return;
    int lane = threadIdx.x & 31;
    int s, d;
    if (e < EE) { s = src[e]; d = dst[e]; }
    else        { s = d = e - EE; }
    const float* xls = xl + (size_t)s * HIDC;
    __builtin_prefetch(xls, 0, 3);
    int cbase = lane * 8;
    int h = cbase / C;
    float alpha = elog[(size_t)e * H + h] / den[(size_t)d * H + h];
    float* out = acc + (size_t)d * HIDC;
    #pragma unroll
    for (int j = 0; j < 8; ++j) {
        int c = cbase + j;
        atomicAdd(&out[c], xls[c] * alpha);
    }
}

// ---------- bias + LayerNorm + ReLU (wave per node) ----------
__global__ void k_ln_relu(const float* __restrict__ acc, const float* __restrict__ bias,
                          const float* __restrict__ g, const float* __restrict__ be,
                          float* __restrict__ out) {
    int i = blockIdx.x * (blockDim.x >> 5) + (threadIdx.x >> 5);
    if (i >= NN) return;
    int lane = threadIdx.x & 31;
    float v[8];
    float s = 0.0f, sq = 0.0f;
    const float* row = acc + (size_t)i * HIDC;
    #pragma unroll
    for (int j = 0; j < 8; ++j) {
        int c = lane * 8 + j;
        v[j] = row[c] + bias[c];
        s += v[j];
        sq += v[j] * v[j];
    }
    #pragma unroll
    for (int off = 16; off >= 1; off >>= 1) {
        s  += __shfl_xor(s,  off, 32);
        sq += __shfl_xor(sq, off, 32);
    }
    float mu = s * (1.0f / HIDC);
    float var = sq * (1.0f / HIDC) - mu * mu;
    float inv = rsqrtf(var + LNEPS);
    float* orow = out + (size_t)i * HIDC;
    #pragma unroll
    for (int j = 0; j < 8; ++j) {
        int c = lane * 8 + j;
        float y = (v[j] - mu) * inv * g[c] + be[c];
        orow[c] = fmaxf(y, 0.0f);
    }
}

// ---------- policy tail: p1[N,128] @ Wp2[128,4] + bp2, global atomic max ----------
__global__ void k_policy_logits(const float* __restrict__ p1, const float* __restrict__ Wp2,
                                const float* __restrict__ bp2, float* __restrict__ plog,
                                unsigned* __restrict__ gmx) {
    int idx = blockIdx.x * blockDim.x + threadIdx.x;
    if (idx >= NN * 4) return;
    int i = idx >> 2, k = idx & 3;
    float sAcc = bp2[k];
    const float* p = p1 + (size_t)i * 128;
    for (int j = 0; j < 128; ++j) sAcc += p[j] * Wp2[j * 4 + k];
    plog[idx] = sAcc;
    atomicMax(gmx, fmap(sAcc));
}
__global__ void k_softmax_den(const float* __restrict__ plog, const unsigned* __restrict__ gmx,
                              float* __restrict__ gden) {
    int idx = blockIdx.x * blockDim.x + threadIdx.x;
    if (idx >= NN * 4) return;
    atomicAdd(gden, __expf(plog[idx] - funmap(*gmx)));
}
__global__ void k_softmax_out(const float* __restrict__ plog, const unsigned* __restrict__ gmx,
                              const float* __restrict__ gden, float* __restrict__ out) {
    int idx = blockIdx.x * blockDim.x + threadIdx.x;
    if (idx >= NN * 4) return;
    out[idx] = __expf(plog[idx] - funmap(*gmx)) / (*gden);
}

// ---------- launch ----------
extern "C" void kernel_launch(void* const* d_in, const int* in_sizes, int n_in,
                              void* d_out, int out_size, void* d_ws, size_t ws_size,
                              hipStream_t stream) {
    const float* x     = (const float*)d_in[0];
    const int*   ei    = (const int*)d_in[1];
    const float* ea    = (const float*)d_in[2];
    const float* Wl1   = (const float*)d_in[3];
    const float* bl1   = (const float*)d_in[4];
    const float* Wr1   = (const float*)d_in[5];
    const float* br1   = (const float*)d_in[6];
    const float* We1   = (const float*)d_in[7];
    const float* att1  = (const float*)d_in[8];
    const float* bias1 = (const float*)d_in[9];
    const float* g1    = (const float*)d_in[10];
    const float* be1   = (const float*)d_in[11];
    const float* Wl2   = (const float*)d_in[12];
    const float* bl2   = (const float*)d_in[13];
    const float* Wr2   = (const float*)d_in[14];
    const float* br2   = (const float*)d_in[15];
    const float* We2   = (const float*)d_in[16];
    const float* att2  = (const float*)d_in[17];
    const float* bias2 = (const float*)d_in[18];
    const float* g2    = (const float*)d_in[19];
    const float* be2   = (const float*)d_in[20];
    const float* Wp1   = (const float*)d_in[21];
    const float* bp1   = (const float*)d_in[22];
    const float* Wp2   = (const float*)d_in[23];
    const float* bp2   = (const float*)d_in[24];
    float* out = (float*)d_out;

    const int* src = ei;
    const int* dst = ei + EE;

    // workspace layout (floats)
    float* W = (float*)d_ws;
    size_t o = 0;
    float*    deg   = W + o; o += NN;
    float*    loopa = W + o; o += NN;
    unsigned* mxu   = (unsigned*)(W + o); o += (size_t)NN * 4;
    float*    den   = W + o; o += (size_t)NN * 4;
    unsigned* gmx   = (unsigned*)(W + o); o += 1;
    float*    gden  = W + o; o += 1;
    float*    accb  = W + o; o += (size_t)NN * HIDC;
    float*    xl    = W + o; o += (size_t)NN * HIDC;
    float*    xr    = W + o; o += (size_t)NN * HIDC;
    float*    hbuf  = W + o; o += (size_t)NN * HIDC;
    float*    elog  = W + o; o += (size_t)EAT * 4;
    float*    p1    = W + o; o += (size_t)NN * 128;
    float*    plog  = W + o; o += (size_t)NN * 4;

    const int T = 256;
    #define GRID(n) (((n) + T - 1) / T)

    // ---- self-loop attrs ----
    k_fill0<<<GRID(2 * NN), T, 0, stream>>>(deg, 2 * NN);                 // deg + loopa
    k_deg_loop<<<GRID(EE), T, 0, stream>>>(dst, ea, deg, loopa);
    k_loop_norm<<<GRID(NN), T, 0, stream>>>(loopa, deg);

    // ---- conv1 ----
    k_fill0<<<GRID(NN * 8 + 2), T, 0, stream>>>((float*)mxu, NN * 8 + 2); // mxu, den, gmx, gden
    k_fill0<<<GRID(NN * HIDC), T, 0, stream>>>(accb, NN * HIDC);
    k_gemm_wmma<4, HIDC><<<dim3(NN / 16), HIDC, 0, stream>>>(x, 4, Wl1, bl1, xl, 0);
    k_gemm_wmma<4, HIDC><<<dim3(NN / 16), HIDC, 0, stream>>>(x, 4, Wr1, br1, xr, 0);
    k_edge_logits<4, 64><<<GRID(EAT * 32), T, 0, stream>>>(xl, xr, We1, src, dst, ea, loopa,
                                                           att1, elog, mxu);
    k_edge_exp<4><<<GRID(EAT * 4), T, 0, stream>>>(dst, elog, mxu, den);
    k_edge_scatter<4, 64><<<GRID(EAT * 32), T, 0, stream>>>(xl, src, dst, elog, den, accb);
    k_ln_relu<<<GRID(NN * 32), T, 0, stream>>>(accb, bias1, g1, be1, hbuf);

    // ---- conv2 ----
    k_fill0<<<GRID(NN * 8 + 2), T, 0, stream>>>((float*)mxu, NN * 8 + 2);
    k_fill0<<<GRID(NN * HIDC), T, 0, stream>>>(accb, NN * HIDC);
    k_gemm_wmma<HIDC, HIDC><<<dim3(NN / 16), HIDC, 0, stream>>>(hbuf, HIDC, Wl2, bl2, xl, 0);
    k_gemm_wmma<HIDC, HIDC><<<dim3(NN / 16), HIDC, 0, stream>>>(hbuf, HIDC, Wr2, br2, xr, 0);
    k_edge_logits<1, 256><<<GRID(EAT * 32), T, 0, stream>>>(xl, xr, We2, src, dst, ea, loopa,
                                                            att2, elog, mxu);
    k_edge_exp<1><<<GRID(EAT), T, 0, stream>>>(dst, elog, mxu, den);
    k_edge_scatter<1, 256><<<GRID(EAT * 32), T, 0, stream>>>(xl, src, dst, elog, den, accb);
    k_ln_relu<<<GRID(NN * 32), T, 0, stream>>>(accb, bias2, g2, be2, hbuf);

    // ---- policy head + global softmax ----
    k_gemm_wmma<HIDC, 128><<<dim3(NN / 16), 128, 0, stream>>>(hbuf, HIDC, Wp1, bp1, p1, 1);
    k_policy_logits<<<GRID(NN * 4), T, 0, stream>>>(p1, Wp2, bp2, plog, gmx);
    k_softmax_den<<<GRID(NN * 4), T, 0, stream>>>(plog, gmx, gden);
    k_softmax_out<<<GRID(NN * 4), T, 0, stream>>>(plog, gmx, gden, out);
    #undef GRID
}